// PitchLoss_16466904613547
// MI455X (gfx1250) — compile-verified
//
#include <hip/hip_runtime.h>
#include <hip/hip_bf16.h>

typedef float v2f __attribute__((ext_vector_type(2)));
typedef float v8f __attribute__((ext_vector_type(8)));

#define B_DIM   64
#define L_DIM   4096
#define N_NOTES 128
#define THREADS 256
#define CHUNK   (L_DIM / THREADS)   /* 16 */
#define THRESH  0.5f

// Full wave32 reduction using one V_WMMA_F32_16X16X4_F32.
// A (16x4 f32, 2 VGPRs/lane): vgpr0 = per-lane partial p, vgpr1 = 0.
// B = all ones (layout-independent). Then D[M,N] = p[M] + p[M+16] for all N.
// C/D layout: lane 0 holds column N=0 for M=0..7 (VGPR r -> M=r), lane 16
// holds M=8..15. Summing the 8 D VGPRs in lane 0 and lane 16 and adding the
// two covers all 16 row-sums => full 32-lane sum, exact in f32.
__device__ __forceinline__ float wave_reduce_f32_wmma(float p) {
  v2f a;  a[0] = p;    a[1] = 0.0f;
  v2f bm; bm[0] = 1.0f; bm[1] = 1.0f;
  v8f c = {};
  v8f d = __builtin_amdgcn_wmma_f32_16x16x4_f32(
      /*neg_a=*/false, a, /*neg_b=*/false, bm,
      /*c_mod=*/(short)0, c, /*reuse_a=*/false, /*reuse_b=*/false);
  float s = d[0] + d[1] + d[2] + d[3] + d[4] + d[5] + d[6] + d[7];
  return __shfl(s, 0, 32) + __shfl(s, 16, 32);
}

// One workgroup per batch row b:
//  - mean over l of gen_f0[b,0,:] and contours[b,0,:]
//  - segment counting: seg = inclusive cumsum(offsets); counts[seg] += onsets (l>=1)
//  - c[b,note] = -(counts[note]) - (note==0 ? offsets[b,0] : 0), masked by note < total_offsets
__global__ __launch_bounds__(THREADS)
void pitchloss_row_kernel(const float* __restrict__ gen_f0,
                          const float* __restrict__ contours,
                          const int*   __restrict__ onsets,
                          const int*   __restrict__ offsets,
                          float* __restrict__ mean_g,
                          float* __restrict__ mean_t,
                          float* __restrict__ c_out) {
  const int b   = blockIdx.x;
  const int tid = threadIdx.x;

  __shared__ unsigned int counts[N_NOTES + 1];
  __shared__ int   sc[THREADS];
  __shared__ float redg[THREADS / 32];
  __shared__ float redt[THREADS / 32];

  for (int i = tid; i <= N_NOTES; i += THREADS) counts[i] = 0u;

  // ---- row sums (coalesced, prefetch ahead) ----
  const float* g = gen_f0   + (size_t)b * L_DIM;          // (B,1,L) channel 0
  const float* t = contours + (size_t)b * 2 * L_DIM;      // (B,2,L) channel 0
  float pg = 0.0f, pt = 0.0f;
  #pragma unroll 4
  for (int i = tid; i < L_DIM; i += THREADS) {
    __builtin_prefetch(g + i + 4 * THREADS, 0, 1);
    __builtin_prefetch(t + i + 4 * THREADS, 0, 1);
    pg += g[i];
    pt += t[i];
  }

  // ---- WMMA wave reduction of partial sums (EXEC all ones here) ----
  float wg = wave_reduce_f32_wmma(pg);
  float wt = wave_reduce_f32_wmma(pt);
  if ((tid & 31) == 0) { redg[tid >> 5] = wg; redt[tid >> 5] = wt; }

  // ---- block-wide inclusive scan of offsets (contiguous 16-elt chunks) ----
  const int* off = offsets + (size_t)b * L_DIM;
  const int* ons = onsets  + (size_t)b * L_DIM;
  const int  base = tid * CHUNK;
  int loc[CHUNK];
  int run = 0;
  #pragma unroll
  for (int j = 0; j < CHUNK; ++j) { run += off[base + j]; loc[j] = run; }
  sc[tid] = run;
  __syncthreads();                       // also covers counts[] zeroing + red writes
  for (int s = 1; s < THREADS; s <<= 1) {
    int v = (tid >= s) ? sc[tid - s] : 0;
    __syncthreads();
    sc[tid] += v;
    __syncthreads();
  }
  const int exc         = sc[tid] - run;        // exclusive prefix of my chunk
  const int n_notes_row = sc[THREADS - 1];      // total offsets in row

  // ---- segment counting (on_valid excludes l==0) ----
  #pragma unroll
  for (int j = 0; j < CHUNK; ++j) {
    const int l   = base + j;
    const int seg = exc + loc[j];               // inclusive cumsum at l
    const int ov  = (l >= 1) ? ons[l] : 0;
    if (ov != 0 && seg <= N_NOTES) atomicAdd(&counts[seg], (unsigned)ov);
  }
  __syncthreads();

  // ---- emit c[b, note] and row means ----
  if (tid < N_NOTES) {
    float cv = -(float)(int)counts[tid];
    if (tid == 0) cv -= (float)off[0];
    cv = (tid < n_notes_row) ? cv : 0.0f;
    c_out[(size_t)b * N_NOTES + tid] = cv;
  }
  if (tid == 0) {
    float sg = 0.0f, st = 0.0f;
    #pragma unroll
    for (int w = 0; w < THREADS / 32; ++w) { sg += redg[w]; st += redt[w]; }
    mean_g[b] = sg * (1.0f / (float)L_DIM);
    mean_t[b] = st * (1.0f / (float)L_DIM);
  }
}

// Final reduction: loss = mean_{note,b} relu(|c*(mg-mt)/(c+1e-6)| - 0.5)
__global__ __launch_bounds__(THREADS)
void pitchloss_final_kernel(const float* __restrict__ mean_g,
                            const float* __restrict__ mean_t,
                            const float* __restrict__ c_in,
                            float* __restrict__ out) {
  const int tid = threadIdx.x;
  __shared__ float dmean[B_DIM];
  __shared__ float red[THREADS / 32];

  if (tid < B_DIM) dmean[tid] = mean_g[tid] - mean_t[tid];
  __syncthreads();

  float acc = 0.0f;
  const int total = B_DIM * N_NOTES;            // c_in laid out [b][note]
  for (int i = tid; i < total; i += THREADS) {
    const int   b  = i >> 7;                    // N_NOTES == 128
    const float cv = c_in[i];
    const float diff = fabsf(cv * dmean[b] / (cv + 1e-6f));
    const float r = diff - THRESH;
    acc += (r > 0.0f) ? r : 0.0f;
  }

  float w = wave_reduce_f32_wmma(acc);
  if ((tid & 31) == 0) red[tid >> 5] = w;
  __syncthreads();
  if (tid == 0) {
    float s = 0.0f;
    #pragma unroll
    for (int i = 0; i < THREADS / 32; ++i) s += red[i];
    out[0] = s / (float)total;
  }
}

extern "C" void kernel_launch(void* const* d_in, const int* in_sizes, int n_in,
                              void* d_out, int out_size, void* d_ws, size_t ws_size,
                              hipStream_t stream) {
  (void)in_sizes; (void)n_in; (void)out_size; (void)ws_size;
  const float* gen_f0   = (const float*)d_in[0];
  const float* contours = (const float*)d_in[1];
  const int*   onsets   = (const int*)d_in[2];
  const int*   offsets  = (const int*)d_in[3];
  // d_in[4] = n_notes_max (scalar 128, compiled in as N_NOTES)

  float* ws     = (float*)d_ws;
  float* mg     = ws;                   // [64]
  float* mt     = ws + B_DIM;           // [64]
  float* c_vals = ws + 2 * B_DIM;       // [64*128]

  pitchloss_row_kernel<<<B_DIM, THREADS, 0, stream>>>(
      gen_f0, contours, onsets, offsets, mg, mt, c_vals);
  pitchloss_final_kernel<<<1, THREADS, 0, stream>>>(
      mg, mt, c_vals, (float*)d_out);
}